// Encoder_61091614818916
// MI455X (gfx1250) — compile-verified
//
#include <hip/hip_runtime.h>
#include <math.h>

// Problem constants (match reference)
#define NN   4096
#define FIN  3000
#define HIDD 256
#define NH   4
#define NC   64
#define EE   65536
#define PHN  128

typedef __attribute__((ext_vector_type(2))) float v2f;
typedef __attribute__((ext_vector_type(8))) float v8f;

// ---------------------------------------------------------------------------
// f32 WMMA GEMM: C[M,N] = A[M,K] @ B[K,N], row-major f32. Requires M % 128 == 0.
// 256 threads = 8 waves (4x2). Block tile 128x64, K chunks of 32 in LDS.
// Each wave owns a 32x32 output tile: 2 A-frags x 2 B-frags -> 4 WMMAs per
// k-step (V_WMMA_F32_16X16X4_F32).
// LDS tiles are stored K-PAIRED: element (k,m) at tile[k/2][m][k&1], so each
// WMMA fragment is one contiguous 8-byte ds_load_b64 straight into the
// operand VGPR pair (no repacking movs, partial dscnt waits possible).
// ---------------------------------------------------------------------------
#define BM 128
#define BN 64
#define BK 32
#define KP (BK / 2)

__global__ __launch_bounds__(256) void k_gemm_wmma(
    const float* __restrict__ A, const float* __restrict__ B,
    float* __restrict__ C, int M, int Nc, int K) {
  __shared__ v2f lA2[KP][BM + 1];
  __shared__ v2f lB2[KP][BN + 1];
  float* lAf = (float*)lA2;
  float* lBf = (float*)lB2;
  const int t    = threadIdx.x;
  const int lane = t & 31;
  const int wv   = t >> 5;
  const int wr   = wv >> 1;               // 0..3  (32-row strip)
  const int wc   = wv & 1;                // 0..1  (32-col strip)
  const int bm   = blockIdx.y * BM;
  const int bn   = blockIdx.x * BN;
  const int mloc = lane & 15;
  const int kh   = lane >> 4;             // which k-pair of the quad (0/1)
  v8f acc00 = {0.f, 0.f, 0.f, 0.f, 0.f, 0.f, 0.f, 0.f};
  v8f acc01 = acc00, acc10 = acc00, acc11 = acc00;

  for (int k0 = 0; k0 < K; k0 += BK) {
    for (int i = t; i < BM * BK; i += 256) {
      int r = i >> 5, c = i & (BK - 1);
      int gc = k0 + c;
      float v = (gc < K) ? A[(size_t)(bm + r) * K + gc] : 0.f;
      lAf[((c >> 1) * (BM + 1) + r) * 2 + (c & 1)] = v;
    }
    for (int i = t; i < BK * BN; i += 256) {
      int r = i >> 6, c = i & (BN - 1);
      int gr = k0 + r, gc = bn + c;
      float v = (gr < K && gc < Nc) ? B[(size_t)gr * Nc + gc] : 0.f;
      lBf[((r >> 1) * (BN + 1) + c) * 2 + (r & 1)] = v;
    }
    __syncthreads();
    const int am0 = wr * 32 + mloc;
    const int am1 = am0 + 16;
    const int nb0 = wc * 32 + mloc;
#pragma unroll
    for (int kk = 0; kk < BK; kk += 4) {
      const int kp = (kk >> 1) + kh;
      v2f a0 = lA2[kp][am0];
      v2f a1 = lA2[kp][am1];
      v2f b0 = lB2[kp][nb0];
      v2f b1 = lB2[kp][nb0 + 16];
      acc00 = __builtin_amdgcn_wmma_f32_16x16x4_f32(false, a0, false, b0,
                                                    (short)0, acc00, false, false);
      acc01 = __builtin_amdgcn_wmma_f32_16x16x4_f32(false, a0, false, b1,
                                                    (short)0, acc01, false, false);
      acc10 = __builtin_amdgcn_wmma_f32_16x16x4_f32(false, a1, false, b0,
                                                    (short)0, acc10, false, false);
      acc11 = __builtin_amdgcn_wmma_f32_16x16x4_f32(false, a1, false, b1,
                                                    (short)0, acc11, false, false);
    }
    __syncthreads();
  }
  const int rb0 = bm + wr * 32 + ((lane >> 4) << 3);
  const int c0  = bn + wc * 32 + mloc;
#pragma unroll
  for (int r = 0; r < 8; r++) {
    int row0 = rb0 + r;
    int row1 = row0 + 16;
    if (c0 < Nc) {
      C[(size_t)row0 * Nc + c0] = acc00[r];
      C[(size_t)row1 * Nc + c0] = acc10[r];
    }
    if (c0 + 16 < Nc) {
      C[(size_t)row0 * Nc + c0 + 16] = acc01[r];
      C[(size_t)row1 * Nc + c0 + 16] = acc11[r];
    }
  }
}

// ---------------------------------------------------------------------------
// CSR construction (deterministic, no float atomics)
// ---------------------------------------------------------------------------
__global__ __launch_bounds__(256) void k_fill_i32(int* p, int v, int n) {
  int i = blockIdx.x * 256 + threadIdx.x;
  if (i < n) p[i] = v;
}

__global__ __launch_bounds__(256) void k_deg(const int* __restrict__ dst,
                                             int* __restrict__ deg) {
  int e = blockIdx.x * 256 + threadIdx.x;
  if (e < EE) atomicAdd(&deg[dst[e]], 1);
}

__global__ __launch_bounds__(256) void k_scan(const int* __restrict__ deg,
                                              int* __restrict__ rowptr) {
  __shared__ int part[256];
  int t = threadIdx.x;
  int base = t * 16;
  int local[16];
  int s = 0;
  for (int i = 0; i < 16; i++) { local[i] = s; s += deg[base + i]; }
  part[t] = s;
  __syncthreads();
  if (t == 0) {
    int acc = 0;
    for (int i = 0; i < 256; i++) { int v = part[i]; part[i] = acc; acc += v; }
    rowptr[NN] = acc;
  }
  __syncthreads();
  int off = part[t];
  for (int i = 0; i < 16; i++) rowptr[base + i] = off + local[i];
}

// Block-per-node ordered compaction: edges appear in CSR in edge-index order.
__global__ __launch_bounds__(256) void k_csr_fill(const int* __restrict__ dst,
                                                  const int* __restrict__ rowptr,
                                                  int* __restrict__ csr) {
  const int n = blockIdx.x;
  const int t = threadIdx.x;
  const int lane = t & 31, w = t >> 5;
  const int base = rowptr[n];
  __shared__ int wcnt[8], woff[8];
  __shared__ int run;
  if (t == 0) run = 0;
  __syncthreads();
  for (int e0 = 0; e0 < EE; e0 += 256) {
    int e = e0 + t;
    bool m = (e < EE) && (dst[e] == n);
    unsigned bm = (unsigned)__ballot(m);          // wave32: low 32 bits
    int pre = __popc(bm & ((1u << lane) - 1u));
    if (lane == 0) wcnt[w] = __popc(bm);
    __syncthreads();
    if (t == 0) {
      int a = run;
      for (int i = 0; i < 8; i++) { woff[i] = a; a += wcnt[i]; }
      run = a;
    }
    __syncthreads();
    if (m) csr[base + woff[w] + pre] = e;
    __syncthreads();
  }
}

// ---------------------------------------------------------------------------
// GAT layer 1 (4 heads x 64 ch)
// ---------------------------------------------------------------------------
__global__ __launch_bounds__(256) void k_scores1(const float* __restrict__ h1,
                                                 const float* __restrict__ asrc,
                                                 const float* __restrict__ adst,
                                                 float* __restrict__ ssrc,
                                                 float* __restrict__ sdst) {
  int i = blockIdx.x * 256 + threadIdx.x;   // over N*NH
  if (i >= NN * NH) return;
  int n = i >> 2, h = i & 3;
  const float* row = h1 + (size_t)n * HIDD + h * NC;
  const float* as = asrc + h * NC;
  const float* ad = adst + h * NC;
  float s1 = 0.f, s2 = 0.f;
  for (int c = 0; c < NC; c++) { float v = row[c]; s1 += v * as[c]; s2 += v * ad[c]; }
  ssrc[i] = s1; sdst[i] = s2;
}

__global__ __launch_bounds__(256) void k_gat1(
    const float* __restrict__ h1, const float* __restrict__ ssrc,
    const float* __restrict__ sdst, const float* __restrict__ ew,
    const int* __restrict__ srcv, const int* __restrict__ rowptr,
    const int* __restrict__ csr, const float* __restrict__ b1v,
    float* __restrict__ alph, float* __restrict__ zout, float* zout2) {
  const int n = blockIdx.x;
  const int t = threadIdx.x;
  const int beg = rowptr[n];
  const int deg = rowptr[n + 1] - beg;
  __shared__ float red[256];
  __shared__ float mh[4], dh[4], sdn[4];
  __shared__ int ce[256], cs[256];
  const int j = t >> 2, h = t & 3;
  if (t < 4) sdn[t] = sdst[n * 4 + t];
  __syncthreads();
  // max per head
  float lm = -3.4e38f;
  for (int jj = j; jj < deg; jj += 64) {
    int e = csr[beg + jj];
    int s = srcv[e];
    float v = ssrc[s * 4 + h] + sdn[h];
    v = v >= 0.f ? v : 0.2f * v;
    lm = fmaxf(lm, v);
  }
  red[t] = lm; __syncthreads();
  for (int st = 32; st >= 1; st >>= 1) {
    if (j < st) red[t] = fmaxf(red[t], red[t + st * 4]);
    __syncthreads();
  }
  if (t < 4) mh[t] = red[t];
  __syncthreads();
  // exp-sum + stash un-normalized alpha
  float ls = 0.f;
  for (int jj = j; jj < deg; jj += 64) {
    int e = csr[beg + jj];
    int s = srcv[e];
    float v = ssrc[s * 4 + h] + sdn[h];
    v = v >= 0.f ? v : 0.2f * v;
    float ex = __expf(v - mh[h]) * ew[e];
    alph[((size_t)e << 2) + h] = ex;
    ls += ex;
    if (h == 0 && jj < 256) { ce[jj] = e; cs[jj] = s; }
  }
  red[t] = ls; __syncthreads();
  for (int st = 32; st >= 1; st >>= 1) {
    if (j < st) red[t] += red[t + st * 4];
    __syncthreads();
  }
  if (t < 4) dh[t] = red[t];
  __syncthreads();
  // aggregate: channel t = h*64 + c
  const int hh = t >> 6;
  float acc = 0.f;
  for (int jj = 0; jj < deg; jj++) {
    int e, s;
    if (jj < 256) { e = ce[jj]; s = cs[jj]; }
    else          { e = csr[beg + jj]; s = srcv[e]; }
    acc += alph[((size_t)e << 2) + hh] * h1[(size_t)s * HIDD + t];
  }
  float o = acc / (dh[hh] + 1e-16f) + b1v[t];
  o = o > 0.f ? o : 0.f;                 // relu (z == emb == hidden_emb)
  zout[(size_t)n * HIDD + t] = o;
  if (zout2) zout2[(size_t)n * HIDD + t] = o;
}

// ---------------------------------------------------------------------------
// GAT layer 2 (1 head x 3000 ch), no relu, writes h to d_out
// ---------------------------------------------------------------------------
__global__ __launch_bounds__(256) void k_scores2(const float* __restrict__ h2,
                                                 const float* __restrict__ asrc,
                                                 const float* __restrict__ adst,
                                                 float* __restrict__ ssrc,
                                                 float* __restrict__ sdst) {
  const int n = blockIdx.x;
  const int t = threadIdx.x;
  __shared__ float red[256];
  float s1 = 0.f, s2 = 0.f;
  for (int c = t; c < FIN; c += 256) {
    float v = h2[(size_t)n * FIN + c];
    s1 += v * asrc[c]; s2 += v * adst[c];
  }
  red[t] = s1; __syncthreads();
  for (int st = 128; st >= 1; st >>= 1) { if (t < st) red[t] += red[t + st]; __syncthreads(); }
  if (t == 0) ssrc[n] = red[0];
  __syncthreads();
  red[t] = s2; __syncthreads();
  for (int st = 128; st >= 1; st >>= 1) { if (t < st) red[t] += red[t + st]; __syncthreads(); }
  if (t == 0) sdst[n] = red[0];
}

__global__ __launch_bounds__(256) void k_gat2(
    const float* __restrict__ h2, const float* __restrict__ ssrc,
    const float* __restrict__ sdst, const float* __restrict__ ew,
    const int* __restrict__ srcv, const int* __restrict__ rowptr,
    const int* __restrict__ csr, const float* __restrict__ b2v,
    float* __restrict__ alph, float* __restrict__ outh) {
  const int n = blockIdx.x;
  const int t = threadIdx.x;
  const int beg = rowptr[n];
  const int deg = rowptr[n + 1] - beg;
  __shared__ float red[256];
  __shared__ float m0s, d0s;
  __shared__ int cs[256];
  __shared__ float cal[256];
  const float sdn = sdst[n];
  float lm = -3.4e38f;
  for (int jj = t; jj < deg; jj += 256) {
    int e = csr[beg + jj];
    float v = ssrc[srcv[e]] + sdn;
    v = v >= 0.f ? v : 0.2f * v;
    lm = fmaxf(lm, v);
  }
  red[t] = lm; __syncthreads();
  for (int st = 128; st >= 1; st >>= 1) { if (t < st) red[t] = fmaxf(red[t], red[t + st]); __syncthreads(); }
  if (t == 0) m0s = red[0];
  __syncthreads();
  float ls = 0.f;
  for (int jj = t; jj < deg; jj += 256) {
    int e = csr[beg + jj];
    int s = srcv[e];
    float v = ssrc[s] + sdn;
    v = v >= 0.f ? v : 0.2f * v;
    float ex = __expf(v - m0s) * ew[e];
    alph[e] = ex;
    ls += ex;
    if (jj < 256) { cs[jj] = s; cal[jj] = ex; }
  }
  red[t] = ls; __syncthreads();
  for (int st = 128; st >= 1; st >>= 1) { if (t < st) red[t] += red[t + st]; __syncthreads(); }
  if (t == 0) d0s = red[0];
  __syncthreads();
  const float inv = 1.0f / (d0s + 1e-16f);
  for (int c = t; c < FIN; c += 256) {
    float acc = 0.f;
    for (int jj = 0; jj < deg; jj++) {
      int s; float a;
      if (jj < 256) { s = cs[jj]; a = cal[jj]; }
      else { int e = csr[beg + jj]; s = srcv[e]; a = alph[e]; }
      acc += a * h2[(size_t)s * FIN + c];
    }
    outh[(size_t)n * FIN + c] = acc * inv + b2v[c];
  }
}

// ---------------------------------------------------------------------------
// Attention pooling
// ---------------------------------------------------------------------------
__global__ __launch_bounds__(256) void k_poolscore(const float* __restrict__ p,
                                                   const float* __restrict__ bv,
                                                   const float* __restrict__ w2,
                                                   const float* __restrict__ b2s,
                                                   float* __restrict__ scores) {
  int n = blockIdx.x * 256 + threadIdx.x;
  if (n >= NN) return;
  float s = 0.f;
  for (int j = 0; j < PHN; j++) {
    float v = p[(size_t)n * PHN + j] + bv[j];
    v = v > 0.f ? v : 0.f;
    s += v * w2[j];
  }
  scores[n] = s + b2s[0];
}

__global__ __launch_bounds__(256) void k_poolwts(const float* __restrict__ scores,
                                                 const int* __restrict__ gm,
                                                 float* __restrict__ wts) {
  const int i = blockIdx.x;
  const int t = threadIdx.x;
  __shared__ float red[256];
  __shared__ float Ms, Ss;
  const int* mrow = gm + (size_t)i * NN;
  float lm = -3.4e38f;
  for (int j = t; j < NN; j += 256) {
    float v = mrow[j] ? scores[j] : -1e9f;
    lm = fmaxf(lm, v);
  }
  red[t] = lm; __syncthreads();
  for (int st = 128; st >= 1; st >>= 1) { if (t < st) red[t] = fmaxf(red[t], red[t + st]); __syncthreads(); }
  if (t == 0) Ms = red[0];
  __syncthreads();
  float ls = 0.f;
  for (int j = t; j < NN; j += 256) {
    float v = mrow[j] ? scores[j] : -1e9f;
    ls += __expf(v - Ms);
  }
  red[t] = ls; __syncthreads();
  for (int st = 128; st >= 1; st >>= 1) { if (t < st) red[t] += red[t + st]; __syncthreads(); }
  if (t == 0) Ss = red[0];
  __syncthreads();
  const float inv = 1.0f / Ss;
  for (int j = t; j < NN; j += 256) {
    float v = mrow[j] ? scores[j] : -1e9f;
    wts[(size_t)i * NN + j] = __expf(v - Ms) * inv;
  }
}

__global__ __launch_bounds__(256) void k_gnorm(float* __restrict__ g) {
  const int n = blockIdx.x;
  const int t = threadIdx.x;
  __shared__ float red[256];
  float v = g[(size_t)n * HIDD + t];
  red[t] = v * v; __syncthreads();
  for (int st = 128; st >= 1; st >>= 1) { if (t < st) red[t] += red[t + st]; __syncthreads(); }
  float nm = fmaxf(sqrtf(red[0]), 1e-12f);
  float x = v / nm;
  g[(size_t)n * HIDD + t] = 1.f / (1.f + __expf(-x));
}

// ---------------------------------------------------------------------------
// Bilinear discriminator: out[n,0]=hpl·(W g), out[n,1]=hmi·(W g), +bb+bias
// ---------------------------------------------------------------------------
__global__ __launch_bounds__(256) void k_disc(const float* __restrict__ g,
                                              const float* __restrict__ hpl,
                                              const float* __restrict__ hmi,
                                              const float* __restrict__ W,
                                              const float* __restrict__ bbp,
                                              const float* __restrict__ biasp,
                                              float* __restrict__ outp) {
  const int n = blockIdx.x;
  const int t = threadIdx.x;
  __shared__ float gl[256];
  __shared__ float red[256];
  gl[t] = g[(size_t)n * HIDD + t];
  __syncthreads();
  const float* wrow = W + (size_t)t * HIDD;
  float tc = 0.f;
  for (int d = 0; d < HIDD; d++) tc += wrow[d] * gl[d];
  float p1 = hpl[(size_t)n * HIDD + t] * tc;
  float p2 = hmi[(size_t)n * HIDD + t] * tc;
  red[t] = p1; __syncthreads();
  for (int st = 128; st >= 1; st >>= 1) { if (t < st) red[t] += red[t + st]; __syncthreads(); }
  float sc1 = red[0];
  __syncthreads();
  red[t] = p2; __syncthreads();
  for (int st = 128; st >= 1; st >>= 1) { if (t < st) red[t] += red[t + st]; __syncthreads(); }
  if (t == 0) {
    float b = bbp[0] + biasp[0];
    outp[(size_t)n * 2 + 0] = sc1 + b;
    outp[(size_t)n * 2 + 1] = red[0] + b;
  }
}

// ---------------------------------------------------------------------------
extern "C" void kernel_launch(void* const* d_in, const int* in_sizes, int n_in,
                              void* d_out, int out_size, void* d_ws, size_t ws_size,
                              hipStream_t stream) {
  (void)in_sizes; (void)n_in; (void)out_size; (void)ws_size;
  const float* feat    = (const float*)d_in[0];
  const float* feat_a  = (const float*)d_in[1];
  const int*   eidx    = (const int*)d_in[2];
  const float* ew      = (const float*)d_in[3];
  const int*   gmask   = (const int*)d_in[4];
  const float* W1      = (const float*)d_in[5];
  const float* a_src1  = (const float*)d_in[6];
  const float* a_dst1  = (const float*)d_in[7];
  const float* b1v     = (const float*)d_in[8];
  const float* W2      = (const float*)d_in[9];
  const float* a_src2  = (const float*)d_in[10];
  const float* a_dst2  = (const float*)d_in[11];
  const float* b2v     = (const float*)d_in[12];
  const float* pa_w1   = (const float*)d_in[13];
  const float* pa_b1   = (const float*)d_in[14];
  const float* pa_w2   = (const float*)d_in[15];
  const float* pa_b2   = (const float*)d_in[16];
  const float* disc_w  = (const float*)d_in[17];
  const float* disc_bb = (const float*)d_in[18];
  const float* disc_bs = (const float*)d_in[19];
  const int* srcv = eidx;
  const int* dstv = eidx + EE;

  // Workspace layout (floats, 256B aligned regions)
  float* wsf = (float*)d_ws;
  size_t off = 0;
  auto alloc = [&](size_t nelem) {
    float* p = wsf + off;
    off += (nelem + 63) & ~(size_t)63;
    return p;
  };
  float* regA  = alloc((size_t)NN * NN);        // h2 [N,FIN] then wts [N,N]
  float* h1    = alloc((size_t)NN * HIDD);
  float* h1a   = alloc((size_t)NN * HIDD);
  float* z     = alloc((size_t)NN * HIDD);
  float* za    = alloc((size_t)NN * HIDD);
  float* ssrc1 = alloc((size_t)NN * NH);
  float* sdst1 = alloc((size_t)NN * NH);
  float* ssrc2 = alloc(NN);
  float* sdst2 = alloc(NN);
  float* alph  = alloc((size_t)EE * NH);
  float* p1    = alloc((size_t)NN * PHN);
  float* p1a   = alloc((size_t)NN * PHN);
  float* scor  = alloc(NN);
  float* scora = alloc(NN);
  float* graw  = alloc((size_t)NN * HIDD);
  float* grawa = alloc((size_t)NN * HIDD);
  int* deg    = (int*)alloc(NN);
  int* rowptr = (int*)alloc(NN + 1);
  int* csr    = (int*)alloc(EE);

  float* out        = (float*)d_out;
  float* out_hidden = out;                                   // [N,HIDD]
  float* out_h      = out + (size_t)NN * HIDD;               // [N,FIN]
  float* out_ret    = out_h + (size_t)NN * FIN;              // [N,2]
  float* out_reta   = out_ret + (size_t)NN * 2;              // [N,2]

  const dim3 blk(256);
  const dim3 g_nodes(NN);
  const int  gy = NN / BM;                                   // M always 4096

  // --- CSR by destination (deterministic) ---
  k_fill_i32<<<dim3((NN + 255) / 256), blk, 0, stream>>>(deg, 0, NN);
  k_deg<<<dim3((EE + 255) / 256), blk, 0, stream>>>(dstv, deg);
  k_scan<<<dim3(1), blk, 0, stream>>>(deg, rowptr);
  k_csr_fill<<<g_nodes, blk, 0, stream>>>(dstv, rowptr, csr);

  // --- branch 1: GAT1(feat) -> z (== hidden_emb == emb) ---
  k_gemm_wmma<<<dim3((HIDD + BN - 1) / BN, gy), blk, 0, stream>>>(feat, W1, h1, NN, HIDD, FIN);
  k_scores1<<<dim3((NN * NH + 255) / 256), blk, 0, stream>>>(h1, a_src1, a_dst1, ssrc1, sdst1);
  k_gat1<<<g_nodes, blk, 0, stream>>>(h1, ssrc1, sdst1, ew, srcv, rowptr, csr, b1v, alph, z, out_hidden);

  // --- GAT2(z) -> h ---
  k_gemm_wmma<<<dim3((FIN + BN - 1) / BN, gy), blk, 0, stream>>>(z, W2, regA, NN, FIN, HIDD);
  k_scores2<<<g_nodes, blk, 0, stream>>>(regA, a_src2, a_dst2, ssrc2, sdst2);
  k_gat2<<<g_nodes, blk, 0, stream>>>(regA, ssrc2, sdst2, ew, srcv, rowptr, csr, b2v, alph, out_h);

  // --- augmented branch: GAT1(feat_a) -> z_a ---
  k_gemm_wmma<<<dim3((HIDD + BN - 1) / BN, gy), blk, 0, stream>>>(feat_a, W1, h1a, NN, HIDD, FIN);
  k_scores1<<<dim3((NN * NH + 255) / 256), blk, 0, stream>>>(h1a, a_src1, a_dst1, ssrc1, sdst1);
  k_gat1<<<g_nodes, blk, 0, stream>>>(h1a, ssrc1, sdst1, ew, srcv, rowptr, csr, b1v, alph, za, nullptr);

  // --- attention pooling (both branches) ---
  k_gemm_wmma<<<dim3((PHN + BN - 1) / BN, gy), blk, 0, stream>>>(z, pa_w1, p1, NN, PHN, HIDD);
  k_gemm_wmma<<<dim3((PHN + BN - 1) / BN, gy), blk, 0, stream>>>(za, pa_w1, p1a, NN, PHN, HIDD);
  k_poolscore<<<dim3((NN + 255) / 256), blk, 0, stream>>>(p1, pa_b1, pa_w2, pa_b2, scor);
  k_poolscore<<<dim3((NN + 255) / 256), blk, 0, stream>>>(p1a, pa_b1, pa_w2, pa_b2, scora);

  k_poolwts<<<g_nodes, blk, 0, stream>>>(scor, gmask, regA);
  k_gemm_wmma<<<dim3(HIDD / BN, gy), blk, 0, stream>>>(regA, z, graw, NN, HIDD, NN);
  k_gnorm<<<g_nodes, blk, 0, stream>>>(graw);

  k_poolwts<<<g_nodes, blk, 0, stream>>>(scora, gmask, regA);
  k_gemm_wmma<<<dim3(HIDD / BN, gy), blk, 0, stream>>>(regA, za, grawa, NN, HIDD, NN);
  k_gnorm<<<g_nodes, blk, 0, stream>>>(grawa);

  // --- discriminator ---
  k_disc<<<g_nodes, blk, 0, stream>>>(graw,  z,  za, disc_w, disc_bb, disc_bs, out_ret);
  k_disc<<<g_nodes, blk, 0, stream>>>(grawa, za, z,  disc_w, disc_bb, disc_bs, out_reta);
}